// KNNGraph_37752762532327
// MI455X (gfx1250) — compile-verified
//
#include <hip/hip_runtime.h>

// Problem constants (match reference setup_inputs()).
#define N_SETS 4
#define M_PTS  4096
#define D_DIM  256
#define K_NN   16

#define ROWS   32            // rows per block: two 16-row WMMA blocks share B frags
#define CHUNK  1024          // columns of the dist tile resident in LDS at once
#define APAD   260           // padded dword stride of A tile (conflict-free frags)
#define CPAD   1028          // padded dword stride of dist tile (16B aligned)

typedef __attribute__((ext_vector_type(2))) float        v2f;
typedef __attribute__((ext_vector_type(8))) float        v8f;
typedef __attribute__((ext_vector_type(4))) unsigned int v4u;
typedef __attribute__((ext_vector_type(4))) int          v4i;
typedef __attribute__((ext_vector_type(8))) int          v8i;

// ---------------------------------------------------------------------------
// Kernel 1: squared row norms  nrm[n*M + m] = sum_d x[n,m,d]^2
// ---------------------------------------------------------------------------
__global__ void knn_norms_kernel(const float* __restrict__ x,
                                 float* __restrict__ nrm) {
  const int row  = blockIdx.x * 8 + (threadIdx.x >> 5);
  const int lane = threadIdx.x & 31;
  const float* p = x + (size_t)row * D_DIM + lane * 8;
  const float4 a = *(const float4*)p;
  const float4 b = *(const float4*)(p + 4);
  float s = a.x*a.x + a.y*a.y + a.z*a.z + a.w*a.w
          + b.x*b.x + b.y*b.y + b.z*b.z + b.w*b.w;
#pragma unroll
  for (int off = 16; off > 0; off >>= 1) s += __shfl_xor(s, off, 32);
  if (lane == 0) nrm[row] = s;
}

// Register-resident ascending sorted insert (constant indices only -> VGPRs).
__device__ __forceinline__ void insert16(float (&bv)[K_NN], int (&bi)[K_NN],
                                         float v, int idx) {
  if (v >= bv[K_NN - 1]) return;
#pragma unroll
  for (int j = K_NN - 1; j >= 1; --j) {
    if (v < bv[j - 1])    { bv[j] = bv[j - 1]; bi[j] = bi[j - 1]; }
    else if (v < bv[j])   { bv[j] = v;         bi[j] = idx;       }
  }
  if (v < bv[0]) { bv[0] = v; bi[0] = idx; }
}

// Wave32 shfl-argmin merge of 32 sorted lists -> emit global top-16 of a row.
__device__ __forceinline__ void merge_row(float (&bv)[K_NN], int (&bi)[K_NN],
                                          int lane, int node, int nodeOff,
                                          int* __restrict__ srcOut,
                                          int* __restrict__ dstOut,
                                          float* __restrict__ distOut) {
#pragma unroll 1
  for (int r = 0; r < K_NN; ++r) {
    float v = bv[0]; int gi = bi[0]; int who = lane;
#pragma unroll
    for (int off = 16; off > 0; off >>= 1) {
      const float vo  = __shfl_xor(v, off, 32);
      const int  gio  = __shfl_xor(gi, off, 32);
      const int  whoo = __shfl_xor(who, off, 32);
      if (vo < v || (vo == v && gio < gi)) { v = vo; gi = gio; who = whoo; }
    }
    const bool won = (who == lane);
#pragma unroll
    for (int j = 0; j < K_NN - 1; ++j)
      if (won) { bv[j] = bv[j + 1]; bi[j] = bi[j + 1]; }
    if (won) bv[K_NN - 1] = 3.402823466e38f;
    if (lane == r) {
      const size_t e = (size_t)node * K_NN + r;
      srcOut[e]  = gi + nodeOff;
      dstOut[e]  = node;
      distOut[e] = v;
    }
  }
}

// ---------------------------------------------------------------------------
// Kernel 2: fused distance tiles (fp32 WMMA, B-frag reuse x2) + top-16.
// Grid: N_SETS * (M_PTS/ROWS) blocks of 512 threads (16 wave32s).
// ---------------------------------------------------------------------------
__global__ __launch_bounds__(512, 1)
void knn_graph_kernel(const float* __restrict__ x,
                      const float* __restrict__ nrm,
                      int*   __restrict__ srcOut,
                      int*   __restrict__ dstOut,
                      float* __restrict__ distOut) {
  __shared__ float aLds[ROWS * APAD];     // 32 x 256 A tile, TDM-padded
  __shared__ float distLds[ROWS * CPAD];  // 32 x 1024 dist tile
  __shared__ float sqA[ROWS];

  const int n       = blockIdx.x >> 7;             // M/ROWS = 128 blocks/set
  const int rowBase = (blockIdx.x & 127) * ROWS;
  const float* xs = x   + (size_t)n * M_PTS * D_DIM;
  const float* nr = nrm + (size_t)n * M_PTS;

  const int t     = threadIdx.x;
  const int w     = t >> 5;                        // wave id 0..15
  const int lane  = t & 31;
  const int m16   = lane & 15;
  const int khalf = lane >> 4;

  // ---- stage A tile: Tensor Data Mover, 2D tile + 4-dword pad / 256 dwords
#if __has_builtin(__builtin_amdgcn_tensor_load_to_lds)
  if (w == 0) {
    const unsigned long long ga =
        (unsigned long long)(size_t)(xs + (size_t)rowBase * D_DIM);
    v4u g0;
    g0.x = 1u;                                     // count=1 valid descriptor
    g0.y = (unsigned int)(size_t)aLds;             // lds_addr (low 32 = offset)
    g0.z = (unsigned int)(ga & 0xFFFFFFFFu);       // global_addr[31:0]
    g0.w = (unsigned int)((ga >> 32) & 0x1FFFFFFu) // global_addr[56:32]
         | (2u << 30);                             // type=2 ("image")
    v8i g1;
    g1[0] = (int)((2u << 16)        // data_size = 4B
                | (1u << 20)        // pad_enable
                | (7u << 22)        // pad_interval: 256 dwords
                | (3u << 25));      // pad_amount: 4 dwords -> stride 260
    g1[1] = (int)(256u << 16);      // tensor_dim0 = 256 (bits 79:48, low part)
    g1[2] = (int)(32u << 16);       // tensor_dim1 = 32  (bits 111:80, low part)
    g1[3] = (int)(256u << 16);      // tile_dim0 = 256   (bits 127:112)
    g1[4] = 32;                     // tile_dim1 = 32    (bits 143:128)
    g1[5] = 256;                    // tensor_dim0_stride = 256 (bits 207:160)
    g1[6] = 0;
    g1[7] = 0;
    v4i zz = {0, 0, 0, 0};
#if __clang_major__ >= 23
    v8i z8 = {0, 0, 0, 0, 0, 0, 0, 0};
    __builtin_amdgcn_tensor_load_to_lds(g0, g1, zz, zz, z8, 0);
#else
    __builtin_amdgcn_tensor_load_to_lds(g0, g1, zz, zz, 0);
#endif
    __builtin_amdgcn_s_wait_tensorcnt(0);
  }
#else
  {
    const int r = t >> 4;                          // row 0..31
    const int c = (t & 15) * 16;                   // 16 floats per thread
    const float* s = xs + (size_t)(rowBase + r) * D_DIM + c;
    float* d = &aLds[r * APAD + c];
#pragma unroll
    for (int q = 0; q < 4; ++q)
      *(float4*)(d + 4 * q) = *(const float4*)(s + 4 * q);
  }
#endif
  if (t < ROWS) sqA[t] = nr[rowBase + t];
  __syncthreads();

  float bv0[K_NN]; int bi0[K_NN];                  // row w
  float bv1[K_NN]; int bi1[K_NN];                  // row w+16
#pragma unroll
  for (int j = 0; j < K_NN; ++j) {
    bv0[j] = 3.402823466e38f; bi0[j] = 0;
    bv1[j] = 3.402823466e38f; bi1[j] = 0;
  }

  for (int chunk = 0; chunk < M_PTS / CHUNK; ++chunk) {
    const int chunkBase = chunk * CHUNK;

    // ---- phase A: two 16-row WMMA blocks share every B fragment ----
#pragma unroll 1
    for (int tt = 0; tt < (CHUNK / 16) / 16; ++tt) {   // 4 col tiles / wave
      const int colLocal0 = (w * ((CHUNK / 16) / 16) + tt) * 16;
      const int col = chunkBase + colLocal0 + m16;
      const float* bptr = xs + (size_t)col * D_DIM;
      v8f c0 = {0.f, 0.f, 0.f, 0.f, 0.f, 0.f, 0.f, 0.f};
      v8f c1 = {0.f, 0.f, 0.f, 0.f, 0.f, 0.f, 0.f, 0.f};
#pragma unroll 8
      for (int kc = 0; kc < D_DIM; kc += 4) {
        const v2f bf = *(const v2f*)(bptr + kc + 2 * khalf);
        const v2f a0 = *(const v2f*)&aLds[m16 * APAD + kc + 2 * khalf];
        const v2f a1 = *(const v2f*)&aLds[(16 + m16) * APAD + kc + 2 * khalf];
        c0 = __builtin_amdgcn_wmma_f32_16x16x4_f32(
                 false, a0, false, bf, (short)0, c0, false, false);
        c1 = __builtin_amdgcn_wmma_f32_16x16x4_f32(
                 false, a1, false, bf, (short)0, c1, false, false);
      }
      const float sqb = nr[col];
#pragma unroll
      for (int j = 0; j < 8; ++j) {
        const int rowj = j + 8 * khalf;
        distLds[rowj * CPAD + colLocal0 + m16] =
            sqA[rowj] + sqb - 2.0f * c0[j];
        distLds[(16 + rowj) * CPAD + colLocal0 + m16] =
            sqA[16 + rowj] + sqb - 2.0f * c1[j];
      }
    }
    __syncthreads();

    // ---- phase B: wave w scans rows w and w+16 (b128 LDS reads) ----
    for (int base = 0; base < CHUNK; base += 128) {
      const int cg = chunkBase + base + 4 * lane;
      const float4 q0 = *(const float4*)&distLds[w * CPAD + base + 4 * lane];
      insert16(bv0, bi0, q0.x, cg);
      insert16(bv0, bi0, q0.y, cg + 1);
      insert16(bv0, bi0, q0.z, cg + 2);
      insert16(bv0, bi0, q0.w, cg + 3);
      const float4 q1 =
          *(const float4*)&distLds[(w + 16) * CPAD + base + 4 * lane];
      insert16(bv1, bi1, q1.x, cg);
      insert16(bv1, bi1, q1.y, cg + 1);
      insert16(bv1, bi1, q1.z, cg + 2);
      insert16(bv1, bi1, q1.w, cg + 3);
    }
    __syncthreads();   // before next chunk overwrites distLds
  }

  // ---- emit edges for the wave's two rows ----
  const int nodeOff = n * M_PTS;
  merge_row(bv0, bi0, lane, nodeOff + rowBase + w,      nodeOff,
            srcOut, dstOut, distOut);
  merge_row(bv1, bi1, lane, nodeOff + rowBase + w + 16, nodeOff,
            srcOut, dstOut, distOut);
}

// ---------------------------------------------------------------------------
extern "C" void kernel_launch(void* const* d_in, const int* in_sizes, int n_in,
                              void* d_out, int out_size, void* d_ws,
                              size_t ws_size, hipStream_t stream) {
  (void)in_sizes; (void)n_in; (void)out_size; (void)ws_size;
  const float* x = (const float*)d_in[0];
  float* nrm = (float*)d_ws;                       // N*M floats = 64 KB

  const int E = N_SETS * M_PTS * K_NN;             // 262144 edges
  int*   srcOut  = (int*)d_out;
  int*   dstOut  = (int*)d_out + E;
  float* distOut = (float*)d_out + 2 * (size_t)E;

  knn_norms_kernel<<<(N_SETS * M_PTS) / 8, 256, 0, stream>>>(x, nrm);
  knn_graph_kernel<<<N_SETS * (M_PTS / ROWS), 512, 0, stream>>>(
      x, nrm, srcOut, dstOut, distOut);
}